// LNNEncoder_40381282517232
// MI455X (gfx1250) — compile-verified
//
#include <hip/hip_runtime.h>

// ---------------------------------------------------------------------------
// LTC (liquid time-constant) sequential scan, UNITS=32 == wave32.
// One wave, one lane per neuron. Latency-bound recurrence: 32768 steps x 6
// unfolds; each unfold = 32 tanh-gated synapse contributions per lane with
// reductions fused into FMAs. Parameters register-resident (4 x 32 VGPRs).
// v2: software-pipelined x[t] load; split num/den accumulator chains.
// ---------------------------------------------------------------------------

#define LTC_UNITS   32
#define LTC_LATENT  20
#define LTC_SEQ     32768
#define LTC_UNFOLDS 6
#define LTC_EPS     1e-8f

__device__ __forceinline__ float ltc_softplus(float x) {
    // inputs are small (|x| <= ~8): no overflow concerns; one-time cost
    return logf(1.0f + expf(x));
}

__device__ __forceinline__ float ltc_tanh(float y) {
#if defined(__has_builtin)
#if __has_builtin(__builtin_amdgcn_tanhf)
    return __builtin_amdgcn_tanhf(y);          // gfx1250 v_tanh_f32: 1 trans op
#else
    // tanh(y) = 1 - 2/(exp2(2y*log2e)+1): 2 trans ops
    float e = __builtin_amdgcn_exp2f(y * 2.885390081777927f);
    return 1.0f - 2.0f * __builtin_amdgcn_rcpf(e + 1.0f);
#endif
#else
    float e = __builtin_amdgcn_exp2f(y * 2.885390081777927f);
    return 1.0f - 2.0f * __builtin_amdgcn_rcpf(e + 1.0f);
#endif
}

__global__ __launch_bounds__(LTC_UNITS, 1)
void ltc_scan_kernel(const float* __restrict__ x,        // [SEQ]
                     const float* __restrict__ gleak,    // [U]
                     const float* __restrict__ vleak,    // [U]
                     const float* __restrict__ cm,       // [U]
                     const float* __restrict__ sigma,    // [U,U]
                     const float* __restrict__ mu,       // [U,U]
                     const float* __restrict__ w,        // [U,U]
                     const float* __restrict__ erev,     // [U,U]
                     const float* __restrict__ smu,      // [1,U]
                     const float* __restrict__ ssigma,   // [1,U]
                     const float* __restrict__ sw,       // [1,U]
                     const float* __restrict__ serev,    // [1,U]
                     const float* __restrict__ mask,     // [U,U]
                     const float* __restrict__ smask,    // [1,U]
                     const float* __restrict__ input_w,  // [1]
                     const float* __restrict__ input_b,  // [1]
                     const float* __restrict__ output_w, // [LATENT]
                     const float* __restrict__ output_b, // [LATENT]
                     float* __restrict__ out)            // [LATENT]
{
    const int j = threadIdx.x;                 // post-neuron owned by this lane

    // ---- one-time parameter transform: column j of each 32x32 matrix -------
    // W*sigmoid(s*(v-m)) = Wh + Wh*tanh(A*v - B), Wh = W/2, A = s/2, B = s*m/2
    float A  [LTC_UNITS];
    float Bc [LTC_UNITS];
    float Wh [LTC_UNITS];
    float WEh[LTC_UNITS];
    float csum_den = 0.0f, csum_num = 0.0f;
#pragma unroll
    for (int i = 0; i < LTC_UNITS; ++i) {
        const int idx = i * LTC_UNITS + j;     // [pre=i][post=j], coalesced in j
        const float W  = ltc_softplus(w[idx]) * mask[idx];
        const float wh = 0.5f * W;
        const float sg = sigma[idx];
        A  [i] = 0.5f * sg;
        Bc [i] = 0.5f * sg * mu[idx];
        Wh [i] = wh;
        WEh[i] = wh * erev[idx];
        csum_den += wh;                        // hoisted constant halves
        csum_num += WEh[i];
    }

    const float cmt = ltc_softplus(cm[j]) * (float)LTC_UNFOLDS;
    const float gl  = ltc_softplus(gleak[j]);
    const float glv = gl * vleak[j];
    const float cge = cmt + gl + LTC_EPS;      // constant part of denominator

    // sensory synapse constants (NFEAT == 1)
    const float s_wh = 0.5f * ltc_softplus(sw[j]) * smask[j];
    const float s_A  = 0.5f * ssigma[j];
    const float s_B  = s_A * smu[j];
    const float s_er = serev[j];
    const float iw = input_w[0];
    const float ib = input_b[0];

    float v  = 0.0f;                           // v0 = zeros
    float xt = x[0];                           // software pipeline: x for step t

#pragma unroll 1
    for (int t = 0; t < LTC_SEQ; ++t) {
        // issue next step's uniform load + prefetch now; its latency hides
        // behind the ~6*160 VALU ops of this step's unfolds
        int tn = t + 1;      if (tn >= LTC_SEQ) tn = LTC_SEQ - 1;
        int tp = t + 128;    if (tp >= LTC_SEQ) tp = LTC_SEQ - 1;
        __builtin_prefetch(&x[tp], 0, 0);      // global_prefetch_b8
        const float xnext = x[tn];

        const float u    = __builtin_fmaf(xt, iw, ib);
        const float st   = ltc_tanh(__builtin_fmaf(s_A, u, -s_B));
        const float sact = __builtin_fmaf(s_wh, st, s_wh);   // s_wh*(1+tanh)
        const float base_den = csum_den + sact;
        const float base_num = __builtin_fmaf(sact, s_er, csum_num);

#pragma unroll 1
        for (int k = 0; k < LTC_UNFOLDS; ++k) {
            // split accumulators: halve the dependent-FMA chain depth
            float num0 = base_num, num1 = 0.0f;
            float den0 = base_den, den1 = 0.0f;
#pragma unroll
            for (int i = 0; i < LTC_UNITS; i += 2) {
                const float vi0 = __int_as_float(
                    __builtin_amdgcn_readlane(__float_as_int(v), i));
                const float vi1 = __int_as_float(
                    __builtin_amdgcn_readlane(__float_as_int(v), i + 1));
                const float t0 = ltc_tanh(__builtin_fmaf(A[i],     vi0, -Bc[i]));
                const float t1 = ltc_tanh(__builtin_fmaf(A[i + 1], vi1, -Bc[i + 1]));
                num0 = __builtin_fmaf(WEh[i],     t0, num0);
                den0 = __builtin_fmaf(Wh[i],      t0, den0);
                num1 = __builtin_fmaf(WEh[i + 1], t1, num1);
                den1 = __builtin_fmaf(Wh[i + 1],  t1, den1);
            }
            const float num = num0 + num1;
            const float den = den0 + den1;
            const float numer = __builtin_fmaf(cmt, v, glv) + num;
            const float d     = cge + den;
            float r = __builtin_amdgcn_rcpf(d);            // fast rcp
            r = r * __builtin_fmaf(-d, r, 2.0f);           // 1 Newton step
            v = numer * r;
        }
        xt = xnext;
    }

    if (j < LTC_LATENT) {
        out[j] = __builtin_fmaf(v, output_w[j], output_b[j]);
    }
}

extern "C" void kernel_launch(void* const* d_in, const int* in_sizes, int n_in,
                              void* d_out, int out_size, void* d_ws, size_t ws_size,
                              hipStream_t stream) {
    (void)in_sizes; (void)n_in; (void)d_ws; (void)ws_size; (void)out_size;
    const float* x        = (const float*)d_in[0];
    const float* gleak    = (const float*)d_in[1];
    const float* vleak    = (const float*)d_in[2];
    const float* cm       = (const float*)d_in[3];
    const float* sigma    = (const float*)d_in[4];
    const float* mu       = (const float*)d_in[5];
    const float* w        = (const float*)d_in[6];
    const float* erev     = (const float*)d_in[7];
    const float* smu      = (const float*)d_in[8];
    const float* ssigma   = (const float*)d_in[9];
    const float* sw       = (const float*)d_in[10];
    const float* serev    = (const float*)d_in[11];
    const float* mask     = (const float*)d_in[12];
    const float* smask    = (const float*)d_in[13];
    const float* input_w  = (const float*)d_in[14];
    const float* input_b  = (const float*)d_in[15];
    const float* output_w = (const float*)d_in[16];
    const float* output_b = (const float*)d_in[17];
    float* out = (float*)d_out;

    ltc_scan_kernel<<<dim3(1), dim3(LTC_UNITS), 0, stream>>>(
        x, gleak, vleak, cm, sigma, mu, w, erev,
        smu, ssigma, sw, serev, mask, smask,
        input_w, input_b, output_w, output_b, out);
}